// TripletLoss_84293028151333
// MI455X (gfx1250) — compile-verified
//
#include <hip/hip_runtime.h>
#include <math.h>

typedef __attribute__((ext_vector_type(2))) float v2f;
typedef __attribute__((ext_vector_type(8))) float v8f;

#define BSZ 512
#define DSZ 256
#define MARGIN_F 0.2f

// ---------------------------------------------------------------------------
// Kernel 1: squared norms of each row (trivial, 131K FLOPs)
// ---------------------------------------------------------------------------
__global__ void sqnorm_k(const float* __restrict__ x, float* __restrict__ sqn) {
    int row = blockIdx.x * blockDim.x + threadIdx.x;
    if (row < BSZ) {
        const float* p = x + row * DSZ;
        float s = 0.f;
        #pragma unroll 8
        for (int k = 0; k < DSZ; ++k) s += p[k] * p[k];
        sqn[row] = s;
    }
}

// ---------------------------------------------------------------------------
// Kernel 2: 16x16 tile of G = X*X^T via V_WMMA_F32_16X16X4_F32, then convert
// to Euclidean distance with the reference's safe-sqrt. One wave per tile.
//
// A 16x4 fragment layout (ISA 7.12.2): lane L holds row (L%16); VGPR0/1 hold
// K = k0 + 2*(L>=16) and +1.  B 4x16 fragment has the identical lane mapping
// for X^T since B[k][n] = X[j0+n][k].  C/D: VGPR r, lane L -> M = r + 8*(L>=16),
// N = L%16.
// ---------------------------------------------------------------------------
__global__ void __launch_bounds__(32) gram_dist_k(const float* __restrict__ x,
                                                  const float* __restrict__ sqn,
                                                  float* __restrict__ dist) {
    const int lane = threadIdx.x;          // 0..31, full wave, EXEC all-1s
    const int rl   = lane & 15;
    const int hs   = lane >> 4;            // K-half select
    const int i0   = blockIdx.y * 16;
    const int j0   = blockIdx.x * 16;

    const float* Ar = x + (i0 + rl) * DSZ + hs * 2;
    const float* Br = x + (j0 + rl) * DSZ + hs * 2;

    v8f c = {0.f, 0.f, 0.f, 0.f, 0.f, 0.f, 0.f, 0.f};
    #pragma unroll 4
    for (int k = 0; k < DSZ; k += 4) {
        v2f a = *(const v2f*)(Ar + k);     // 8B aligned (k even)
        v2f b = *(const v2f*)(Br + k);
        c = __builtin_amdgcn_wmma_f32_16x16x4_f32(
                /*neg_a=*/false, a, /*neg_b=*/false, b,
                /*c_mod=*/(short)0, c, /*reuse_a=*/false, /*reuse_b=*/false);
    }

    const int col = j0 + rl;
    const float sc = sqn[col];
    const int rowbase = i0 + hs * 8;
    #pragma unroll
    for (int r = 0; r < 8; ++r) {
        int row  = rowbase + r;
        float sq = sqn[row] + sc - 2.0f * c[r];
        float d  = (sq > 0.f) ? sqrtf(sq) : 0.f;   // safe sqrt, matches ref
        dist[row * BSZ + col] = d;
    }
}

// ---------------------------------------------------------------------------
// Kernel 3: per-anchor triplet accumulation. One block per anchor row i.
// Compact positives/negatives (deterministic single-thread compaction from
// LDS), then an unmasked |P|x|N| relu loop, fixed-order tree reduction.
// No atomics -> bitwise deterministic.
// ---------------------------------------------------------------------------
__global__ void __launch_bounds__(256) triplet_k(const int* __restrict__ labels,
                                                 const float* __restrict__ dist,
                                                 float* __restrict__ tot,
                                                 float* __restrict__ cnt) {
    __shared__ float drow[BSZ];
    __shared__ int   labs[BSZ];
    __shared__ float posd[BSZ];
    __shared__ float negd[BSZ];
    __shared__ int   counts[2];
    __shared__ float red[256];

    const int i = blockIdx.x;
    const int t = threadIdx.x;

    for (int j = t; j < BSZ; j += 256) {
        drow[j] = dist[i * BSZ + j];
        labs[j] = labels[j];
    }
    __syncthreads();

    if (t == 0) {
        const int li = labs[i];
        int np = 0, nn = 0;
        for (int j = 0; j < BSZ; ++j) {
            int l = labs[j];
            if (l == li) { if (j != i) posd[np++] = drow[j]; }
            else         { negd[nn++] = drow[j]; }
        }
        counts[0] = np;
        counts[1] = nn;
    }
    __syncthreads();

    const int np = counts[0];
    const int nn = counts[1];
    const int total = np * nn;

    float s = 0.f;
    for (int p = t; p < total; p += 256) {
        int j = p / nn;
        int k = p - j * nn;
        float v = posd[j] - negd[k] + MARGIN_F;
        s += (v > 0.f) ? v : 0.f;
    }
    red[t] = s;
    __syncthreads();
    for (int off = 128; off > 0; off >>= 1) {
        if (t < off) red[t] += red[t + off];
        __syncthreads();
    }
    if (t == 0) {
        tot[i] = red[0];
        cnt[i] = (float)total;   // <= ~8M, exact in fp32
    }
}

// ---------------------------------------------------------------------------
// Kernel 4: fixed-order final reduction over 512 rows + mean.
// ---------------------------------------------------------------------------
__global__ void finalize_k(const float* __restrict__ tot,
                           const float* __restrict__ cnt,
                           float* __restrict__ out) {
    if (threadIdx.x == 0 && blockIdx.x == 0) {
        float T = 0.f, C = 0.f;
        for (int i = 0; i < BSZ; ++i) { T += tot[i]; C += cnt[i]; }
        out[0] = (C > 0.f) ? (T / fmaxf(C, 1.f)) : 0.f;
    }
}

// ---------------------------------------------------------------------------
extern "C" void kernel_launch(void* const* d_in, const int* in_sizes, int n_in,
                              void* d_out, int out_size, void* d_ws, size_t ws_size,
                              hipStream_t stream) {
    const float* x      = (const float*)d_in[0];   // [512, 256] fp32
    const int*   labels = (const int*)d_in[1];     // [512] int

    float* wsf  = (float*)d_ws;
    float* tot  = wsf;                 // 512
    float* cnt  = wsf + 512;           // 512
    float* sqn  = wsf + 1024;          // 512
    float* dist = wsf + 1536;          // 512*512

    sqnorm_k<<<2, 256, 0, stream>>>(x, sqn);
    gram_dist_k<<<dim3(32, 32), 32, 0, stream>>>(x, sqn, dist);
    triplet_k<<<BSZ, 256, 0, stream>>>(labels, dist, tot, cnt);
    finalize_k<<<1, 32, 0, stream>>>(tot, cnt, (float*)d_out);
}